// HFMoCLlamaMLP_33380485825326
// MI455X (gfx1250) — compile-verified
//
#include <hip/hip_runtime.h>
#include <cstdint>

// Problem constants (from the reference)
constexpr int HDIM = 4096;    // hidden
constexpr int IDIM = 11008;   // intermediate
constexpr int KTOP = 2752;    // kept channels per token

typedef __attribute__((ext_vector_type(2))) float v2f;
typedef __attribute__((ext_vector_type(8))) float v8f;

// ---------------------------------------------------------------------------
// Async copy of 16 bytes per lane from global to LDS (CDNA5 async path,
// tracked by ASYNCcnt). ldsOff is the raw LDS byte address (VDST contents).
// `ldsDst` is the generic pointer to the destination region: it is unused in
// the template but makes the LDS array ESCAPE into the asm, so the compiler
// must assume the asm writes it (otherwise loads from a never-stored shared
// array fold to undef -- observed in the previous round's disassembly).
// ---------------------------------------------------------------------------
__device__ __forceinline__ void async_copy16(uint32_t ldsOff, const float* gsrc,
                                             float* ldsDst) {
    asm volatile("global_load_async_to_lds_b128 %0, %1, off"
                 :: "v"(ldsOff), "v"(gsrc), "v"(ldsDst)
                 : "memory");
}
__device__ __forceinline__ void wait_async0() {
    asm volatile("s_wait_asynccnt 0x0" ::: "memory");
}

// ---------------------------------------------------------------------------
// Kernel 1: z = silu(x @ Wg^T) * (x @ Wu^T)     x:[T,H]  Wg,Wu:[I,H]  z:[T,I]
// Block = 256 threads = 8 waves; tile = 64 tokens x 128 intermediate cols.
// A tile (64 tokens x 16 k) is async-staged into LDS once per block per chunk
// (double buffered), shared by all 8 waves; weights stream from global with
// prefetch. Each wave: 16 cols x 4 token tiles for both g and u.
// ---------------------------------------------------------------------------
__global__ __launch_bounds__(256)
void gateup_kernel(const float* __restrict__ x,
                   const float* __restrict__ Wg,
                   const float* __restrict__ Wu,
                   float* __restrict__ zout) {
    __shared__ float As[2][64 * 16];   // 2 x 4KB double buffer (LDS offset 0)

    const int tid    = threadIdx.x;
    const int lane   = tid & 31;
    const int wave   = tid >> 5;
    const int lane16 = lane & 15;
    const int hi     = lane >> 4;          // half-wave select
    const int ko     = 2 * hi;             // K offset within 4-wide step

    const int col     = blockIdx.x * 128 + wave * 16 + lane16; // intermediate ch
    const int tokBase = blockIdx.y * 64;

    // staging assignment: 4 threads per token row, 16B each
    const int srow = tid >> 2;
    const int sseg = (tid & 3) * 4;
    const float* __restrict__ xsrc = x + (size_t)(tokBase + srow) * HDIM + sseg;
    const uint32_t myLds = (uint32_t)(tid * 16);   // byte offset within a buffer

    const float* __restrict__ grow = Wg + (size_t)col * HDIM;
    const float* __restrict__ urow = Wu + (size_t)col * HDIM;

    v8f accg[4] = {{}, {}, {}, {}};
    v8f accu[4] = {{}, {}, {}, {}};

    constexpr int NCH = HDIM / 16;         // 256 chunks
    async_copy16(myLds, xsrc, &As[0][0]);  // stage chunk 0 into buffer 0

    for (int chunk = 0; chunk < NCH; ++chunk) {
        const int k0  = chunk * 16;
        const int cur = chunk & 1;

        wait_async0();                      // my copies of chunk `chunk` landed
        __syncthreads();                    // all copies landed; prev compute done

        if (chunk + 1 < NCH)                // stage next chunk into free buffer
            async_copy16((uint32_t)((cur ^ 1) * 4096) + myLds,
                         xsrc + (size_t)(k0 + 16), &As[cur ^ 1][0]);

        // stream weights ahead (lowers to global_prefetch_b8)
        __builtin_prefetch(grow + k0 + 64 + ko, 0, 3);
        __builtin_prefetch(urow + k0 + 64 + ko, 0, 3);

#pragma unroll
        for (int kk = 0; kk < 16; kk += 4) {
            const int k = k0 + kk;
            // B (4x16, f32): lane16 = output column; K pair contiguous in H
            v2f bg = *(const v2f*)(grow + k + ko);
            v2f bu = *(const v2f*)(urow + k + ko);
#pragma unroll
            for (int t = 0; t < 4; ++t) {
                // A (16x4, f32) from LDS: row = t*16+lane16, K pair = kk+ko
                v2f a = *(const v2f*)&As[cur][(t * 16 + lane16) * 16 + kk + ko];
                accg[t] = __builtin_amdgcn_wmma_f32_16x16x4_f32(
                    false, a, false, bg, (short)0, accg[t], false, false);
                accu[t] = __builtin_amdgcn_wmma_f32_16x16x4_f32(
                    false, a, false, bu, (short)0, accu[t], false, false);
            }
        }
    }

    // D layout: VGPR r -> row (r + 8*hi), col = lane16
#pragma unroll
    for (int t = 0; t < 4; ++t) {
#pragma unroll
        for (int r = 0; r < 8; ++r) {
            float g = accg[t][r];
            float u = accu[t][r];
            float s = g / (1.0f + __expf(-g));      // silu(g)
            int tok = tokBase + t * 16 + r + 8 * hi;
            zout[(size_t)tok * IDIM + col] = s * u;
        }
    }
}

// ---------------------------------------------------------------------------
// Kernel 2: per-token threshold = K-th largest |z| via 4-pass byte radix
// select on the fp32 bit pattern of |z| (monotone for non-negative floats).
// One block per token; keys staged once in LDS (44KB of the 320KB WGP pool).
// ---------------------------------------------------------------------------
__global__ __launch_bounds__(256)
void topk_threshold_kernel(const float* __restrict__ z,
                           uint32_t* __restrict__ thr) {
    __shared__ uint32_t keys[IDIM];
    __shared__ int      hist[256];
    __shared__ int      s_chosen;
    __shared__ int      s_remaining;

    const int t = blockIdx.x;
    const float* __restrict__ zt = z + (size_t)t * IDIM;

    for (int i = threadIdx.x; i < IDIM; i += 256)
        keys[i] = __float_as_uint(fabsf(zt[i]));
    if (threadIdx.x == 0) s_remaining = KTOP;
    __syncthreads();

    uint32_t prefix = 0, pmask = 0;
#pragma unroll
    for (int shift = 24; shift >= 0; shift -= 8) {
        hist[threadIdx.x & 255] = 0;
        __syncthreads();
        for (int i = threadIdx.x; i < IDIM; i += 256) {
            uint32_t k = keys[i];
            if ((k & pmask) == prefix)
                atomicAdd(&hist[(k >> shift) & 255], 1);
        }
        __syncthreads();
        if (threadIdx.x == 0) {
            int rem = s_remaining;
            int c = 0, b = 255;
            for (; b > 0; --b) {
                c += hist[b];
                if (c >= rem) break;
            }
            if (c < rem) c += hist[0];          // b == 0 fallthrough
            s_chosen    = b;
            s_remaining = rem - (c - hist[b]);
        }
        __syncthreads();
        prefix |= ((uint32_t)s_chosen) << shift;
        pmask  |= 0xFFu << shift;
        __syncthreads();
    }

    if (threadIdx.x == 0) thr[t] = prefix;
}

// ---------------------------------------------------------------------------
// Kernel 3: out = (z masked to top-K by threshold) @ Wd^T
// z:[T,I]  Wd:[H,I]  out:[T,H]. z tile async-staged to LDS (double buffered);
// mask applied with selects after the LDS read (no divergence -> EXEC all 1s).
// ---------------------------------------------------------------------------
__global__ __launch_bounds__(256)
void down_kernel(const float* __restrict__ z,
                 const uint32_t* __restrict__ thr,
                 const float* __restrict__ Wd,
                 float* __restrict__ out) {
    __shared__ float As[2][64 * 16];   // 2 x 4KB double buffer (LDS offset 0)

    const int tid    = threadIdx.x;
    const int lane   = tid & 31;
    const int wave   = tid >> 5;
    const int lane16 = lane & 15;
    const int hi     = lane >> 4;
    const int ko     = 2 * hi;

    const int col     = blockIdx.x * 128 + wave * 16 + lane16; // output h
    const int tokBase = blockIdx.y * 64;

    const int srow = tid >> 2;
    const int sseg = (tid & 3) * 4;
    const float* __restrict__ zsrc = z + (size_t)(tokBase + srow) * IDIM + sseg;
    const uint32_t myLds = (uint32_t)(tid * 16);

    uint32_t tkey[4];
#pragma unroll
    for (int t = 0; t < 4; ++t)
        tkey[t] = thr[tokBase + t * 16 + lane16];

    const float* __restrict__ wrow = Wd + (size_t)col * IDIM;

    v8f acc[4] = {{}, {}, {}, {}};

    constexpr int NCH = IDIM / 16;          // 688 chunks
    async_copy16(myLds, zsrc, &As[0][0]);   // stage chunk 0 into buffer 0

    for (int chunk = 0; chunk < NCH; ++chunk) {
        const int k0  = chunk * 16;
        const int cur = chunk & 1;

        wait_async0();
        __syncthreads();

        if (chunk + 1 < NCH)
            async_copy16((uint32_t)((cur ^ 1) * 4096) + myLds,
                         zsrc + (size_t)(k0 + 16), &As[cur ^ 1][0]);

        __builtin_prefetch(wrow + k0 + 64 + ko, 0, 3);

#pragma unroll
        for (int kk = 0; kk < 16; kk += 4) {
            const int k = k0 + kk;
            v2f b = *(const v2f*)(wrow + k + ko);
#pragma unroll
            for (int t = 0; t < 4; ++t) {
                v2f a = *(const v2f*)&As[cur][(t * 16 + lane16) * 16 + kk + ko];
                uint32_t b0 = __float_as_uint(fabsf(a.x));
                uint32_t b1 = __float_as_uint(fabsf(a.y));
                a.x = (b0 >= tkey[t]) ? a.x : 0.0f;
                a.y = (b1 >= tkey[t]) ? a.y : 0.0f;
                acc[t] = __builtin_amdgcn_wmma_f32_16x16x4_f32(
                    false, a, false, b, (short)0, acc[t], false, false);
            }
        }
    }

#pragma unroll
    for (int t = 0; t < 4; ++t) {
#pragma unroll
        for (int r = 0; r < 8; ++r) {
            int tok = tokBase + t * 16 + r + 8 * hi;
            out[(size_t)tok * HDIM + col] = acc[t][r];
        }
    }
}

// ---------------------------------------------------------------------------
extern "C" void kernel_launch(void* const* d_in, const int* in_sizes, int n_in,
                              void* d_out, int out_size, void* d_ws, size_t ws_size,
                              hipStream_t stream) {
    const float* x  = (const float*)d_in[0];
    const float* Wg = (const float*)d_in[1];
    const float* Wu = (const float*)d_in[2];
    const float* Wd = (const float*)d_in[3];
    float* out = (float*)d_out;

    const int T = in_sizes[0] / HDIM;   // B*S tokens (8192)

    float*    zbuf   = (float*)d_ws;                 // [T, IDIM] fp32
    size_t    zbytes = (size_t)T * IDIM * sizeof(float);
    uint32_t* thr    = (uint32_t*)((char*)d_ws + zbytes); // [T] thresholds

    dim3 blk(256);
    // x = col block (fastest-varying) -> concurrent blocks share x tiles in L2
    gateup_kernel<<<dim3(IDIM / 128, T / 64), blk, 0, stream>>>(x, Wg, Wu, zbuf);
    topk_threshold_kernel<<<dim3(T), blk, 0, stream>>>(zbuf, thr);
    down_kernel<<<dim3(HDIM / 128, T / 64), blk, 0, stream>>>(zbuf, thr, Wd, out);
}